// GNNML1_64991445123425
// MI455X (gfx1250) — compile-verified
//
#include <hip/hip_runtime.h>
#include <hip/hip_bf16.h>

#define NNODES 100000
#define NEDGES 1200000
#define NINF   3
#define NOUTF  64
#define NPG    75
#define NGRAPH ((NNODES + NPG - 1) / NPG)   // 1334
#define EPSBN  1e-5f

typedef __attribute__((ext_vector_type(2))) float v2f;
typedef __attribute__((ext_vector_type(8))) float v8f;

// ---------------------------------------------------------------- utilities

__global__ void zero_kernel(float* __restrict__ p, int n) {
    int t = blockIdx.x * blockDim.x + threadIdx.x;
    if (t < n) p[t] = 0.0f;
}

// ------------------------------------------------- edge scatter (segment_sum)

__global__ void scatter3_kernel(const float* __restrict__ x,
                                const int* __restrict__ src,
                                const int* __restrict__ dst,
                                float* __restrict__ agg) {
    int e = blockIdx.x * blockDim.x + threadIdx.x;
    if (e >= NEDGES) return;
    int s = src[e], d = dst[e];
#pragma unroll
    for (int i = 0; i < 3; ++i)
        atomicAdd(&agg[d * 3 + i], x[s * 3 + i]);
}

__global__ void scatter64_kernel(const float* __restrict__ h,
                                 const int* __restrict__ src,
                                 const int* __restrict__ dst,
                                 float* __restrict__ agg) {
    long long t = (long long)blockIdx.x * blockDim.x + threadIdx.x;
    if (t >= (long long)NEDGES * NOUTF) return;
    int e = (int)(t >> 6);
    int f = (int)(t & 63);
    atomicAdd(&agg[dst[e] * NOUTF + f], h[src[e] * NOUTF + f]);
}

// ------------------------------------------------------ layer 1 (K=3, VALU)

__global__ void layer1_kernel(const float* __restrict__ x,
                              const float* __restrict__ agg,
                              const float* __restrict__ Wc, const float* __restrict__ bc,
                              const float* __restrict__ Wa, const float* __restrict__ ba,
                              const float* __restrict__ Wg1, const float* __restrict__ bg1,
                              const float* __restrict__ Wg2, const float* __restrict__ bg2,
                              float* __restrict__ out) {
    int t = blockIdx.x * blockDim.x + threadIdx.x;
    if (t >= NNODES * NOUTF) return;
    int n = t >> 6, f = t & 63;
    float x0 = x[n * 3 + 0], x1 = x[n * 3 + 1], x2 = x[n * 3 + 2];
    float a0 = agg[n * 3 + 0], a1 = agg[n * 3 + 1], a2 = agg[n * 3 + 2];
    float conv = bc[f] + a0 * Wc[f] + a1 * Wc[64 + f] + a2 * Wc[128 + f];
    float av   = ba[f] + x0 * Wa[f] + x1 * Wa[64 + f] + x2 * Wa[128 + f];
    float g1   = bg1[f] + x0 * Wg1[f] + x1 * Wg1[64 + f] + x2 * Wg1[128 + f];
    float g2   = bg2[f] + x0 * Wg2[f] + x1 * Wg2[64 + f] + x2 * Wg2[128 + f];
    float v = av + conv + g1 * g2;
    out[t] = v > 0.0f ? v : 0.0f;
}

// ------------------------------------- layers 2/3: fused WMMA f32 16x16x4

// One wave computes a 16-node tile across all 64 output features.
// acc = (ba+bc) + x@Wa + agg@Wc ; g1 = bg1 + x@Wg1 ; g2 = bg2 + x@Wg2
// out = relu(acc + g1*g2)
__global__ void __launch_bounds__(256) layer_wmma_kernel(
    const float* __restrict__ X, const float* __restrict__ AGG,
    const float* __restrict__ Wa,  const float* __restrict__ ba,
    const float* __restrict__ Wc,  const float* __restrict__ bc,
    const float* __restrict__ Wg1, const float* __restrict__ bg1,
    const float* __restrict__ Wg2, const float* __restrict__ bg2,
    float* __restrict__ OUT) {

    int wave = (blockIdx.x * blockDim.x + threadIdx.x) >> 5;
    int base = wave * 16;
    if (base >= NNODES) return;          // uniform per wave: EXEC stays all-1s

    int lane = threadIdx.x & 31;
    int l15  = lane & 15;                // A-frag row M  /  B-frag & D col N
    int half = lane >> 4;                // K-half selector

    // Hoist A-fragments for x and agg: lane l holds row (base+l15),
    // K slice = 4k + 2*half .. +1  (ISA 16x4 f32 A layout).
    const float* xrow = X   + (base + l15) * NOUTF;
    const float* arow = AGG + (base + l15) * NOUTF;
    v2f xf[16], gf[16];
#pragma unroll
    for (int k = 0; k < 16; ++k) {
        xf[k] = *(const v2f*)(xrow + 4 * k + 2 * half);
        gf[k] = *(const v2f*)(arow + 4 * k + 2 * half);
    }

#pragma unroll
    for (int ct = 0; ct < 4; ++ct) {
        int cb = ct * 16;
        float bs = ba[cb + l15] + bc[cb + l15];
        v8f acc = {bs, bs, bs, bs, bs, bs, bs, bs};
        float b1 = bg1[cb + l15];
        v8f g1 = {b1, b1, b1, b1, b1, b1, b1, b1};
        float b2 = bg2[cb + l15];
        v8f g2 = {b2, b2, b2, b2, b2, b2, b2, b2};

#pragma unroll
        for (int k = 0; k < 16; ++k) {
            int ka = 4 * k + 2 * half;
            v2f bf;                       // B frag: rows K=ka, ka+1 ; col cb+l15
            bf.x = Wa[ka * NOUTF + cb + l15];
            bf.y = Wa[(ka + 1) * NOUTF + cb + l15];
            acc = __builtin_amdgcn_wmma_f32_16x16x4_f32(
                false, xf[k], false, bf, (short)0, acc, false, false);
        }
#pragma unroll
        for (int k = 0; k < 16; ++k) {
            int ka = 4 * k + 2 * half;
            v2f bf;
            bf.x = Wc[ka * NOUTF + cb + l15];
            bf.y = Wc[(ka + 1) * NOUTF + cb + l15];
            acc = __builtin_amdgcn_wmma_f32_16x16x4_f32(
                false, gf[k], false, bf, (short)0, acc, false, false);
        }
#pragma unroll
        for (int k = 0; k < 16; ++k) {
            int ka = 4 * k + 2 * half;
            v2f bf;
            bf.x = Wg1[ka * NOUTF + cb + l15];
            bf.y = Wg1[(ka + 1) * NOUTF + cb + l15];
            g1 = __builtin_amdgcn_wmma_f32_16x16x4_f32(
                false, xf[k], false, bf, (short)0, g1, false, false);
        }
#pragma unroll
        for (int k = 0; k < 16; ++k) {
            int ka = 4 * k + 2 * half;
            v2f bf;
            bf.x = Wg2[ka * NOUTF + cb + l15];
            bf.y = Wg2[(ka + 1) * NOUTF + cb + l15];
            g2 = __builtin_amdgcn_wmma_f32_16x16x4_f32(
                false, xf[k], false, bf, (short)0, g2, false, false);
        }

        // D layout: VGPR r, lane -> row (r + 8*half), col (cb + l15)
#pragma unroll
        for (int r = 0; r < 8; ++r) {
            float v = acc[r] + g1[r] * g2[r];
            OUT[(base + r + 8 * half) * NOUTF + cb + l15] = v > 0.0f ? v : 0.0f;
        }
    }
}

// ------------------------------------------------- mean pool + BatchNorm(eval)

__global__ void pool_bn_kernel(const float* __restrict__ h,
                               const float* __restrict__ gamma,
                               const float* __restrict__ beta,
                               const float* __restrict__ mean,
                               const float* __restrict__ var,
                               float* __restrict__ pooled) {
    int g = blockIdx.x;
    int f = threadIdx.x;
    int n0 = g * NPG;
    int cnt = NNODES - n0;
    if (cnt > NPG) cnt = NPG;
    float s = 0.0f;
    for (int i = 0; i < cnt; ++i)
        s += h[(n0 + i) * NOUTF + f];
    float p = s / (float)cnt;
    float z = (p - mean[f]) * rsqrtf(var[f] + EPSBN) * gamma[f] + beta[f];
    pooled[g * NOUTF + f] = z;
}

// --------------------------------------------------- classifier head per graph

__global__ void head_kernel(const float* __restrict__ pooled,
                            const float* __restrict__ Wf1, const float* __restrict__ bf1,
                            const float* __restrict__ Wf2, const float* __restrict__ bf2,
                            float* __restrict__ out) {
    __shared__ float zsh[64];
    __shared__ float hsh[32];
    __shared__ float lsh[10];
    __shared__ float lse;
    int g = blockIdx.x;
    int t = threadIdx.x;                 // 32 threads

    zsh[t]      = pooled[g * NOUTF + t];
    zsh[t + 32] = pooled[g * NOUTF + t + 32];
    __syncthreads();

    float s = bf1[t];
    for (int i = 0; i < 64; ++i)
        s += zsh[i] * Wf1[i * 32 + t];
    hsh[t] = s > 0.0f ? s : 0.0f;
    __syncthreads();

    if (t < 10) {
        float l = bf2[t];
        for (int i = 0; i < 32; ++i)
            l += hsh[i] * Wf2[i * 10 + t];
        lsh[t] = l;
    }
    __syncthreads();

    if (t == 0) {
        float m = lsh[0];
        for (int i = 1; i < 10; ++i) m = lsh[i] > m ? lsh[i] : m;
        float se = 0.0f;
        for (int i = 0; i < 10; ++i) se += expf(lsh[i] - m);
        lse = m + logf(se);
    }
    __syncthreads();

    if (t < 10) out[g * 10 + t] = lsh[t] - lse;
}

// -------------------------------------------------------------------- launch

extern "C" void kernel_launch(void* const* d_in, const int* in_sizes, int n_in,
                              void* d_out, int out_size, void* d_ws, size_t ws_size,
                              hipStream_t stream) {
    (void)in_sizes; (void)n_in; (void)out_size; (void)ws_size;

    const float* x   = (const float*)d_in[0];
    const int*   ei  = (const int*)d_in[1];
    const int*   src = ei;
    const int*   dst = ei + NEDGES;
    // d_in[2] = batch (implied by n/NPG, unused)
    const float* Wc1 = (const float*)d_in[3];  const float* bc1 = (const float*)d_in[4];
    const float* W11 = (const float*)d_in[5];  const float* b11 = (const float*)d_in[6];
    const float* W12 = (const float*)d_in[7];  const float* b12 = (const float*)d_in[8];
    const float* W13 = (const float*)d_in[9];  const float* b13 = (const float*)d_in[10];
    const float* Wc2 = (const float*)d_in[11]; const float* bc2 = (const float*)d_in[12];
    const float* W21 = (const float*)d_in[13]; const float* b21 = (const float*)d_in[14];
    const float* W22 = (const float*)d_in[15]; const float* b22 = (const float*)d_in[16];
    const float* W23 = (const float*)d_in[17]; const float* b23 = (const float*)d_in[18];
    const float* Wc3 = (const float*)d_in[19]; const float* bc3 = (const float*)d_in[20];
    const float* W31 = (const float*)d_in[21]; const float* b31 = (const float*)d_in[22];
    const float* W32 = (const float*)d_in[23]; const float* b32 = (const float*)d_in[24];
    const float* W33 = (const float*)d_in[25]; const float* b33 = (const float*)d_in[26];
    const float* bn_g = (const float*)d_in[27]; const float* bn_b = (const float*)d_in[28];
    const float* bn_m = (const float*)d_in[29]; const float* bn_v = (const float*)d_in[30];
    const float* Wf1 = (const float*)d_in[31]; const float* bf1 = (const float*)d_in[32];
    const float* Wf2 = (const float*)d_in[33]; const float* bf2 = (const float*)d_in[34];

    float* ws   = (float*)d_ws;
    float* bufA = ws;                                  // [N,64]
    float* bufB = ws + (size_t)NNODES * NOUTF;         // [N,64]
    float* bufC = ws + (size_t)NNODES * NOUTF * 2;     // [N,64] agg scratch
    float* pooled = ws + (size_t)NNODES * NOUTF * 3;   // [G,64]
    float* out  = (float*)d_out;

    const int n3   = NNODES * 3;
    const int n64  = NNODES * NOUTF;
    const long long e64 = (long long)NEDGES * NOUTF;
    const int tiles = (NNODES + 15) / 16;              // 6250
    const int wmmaBlocks = (tiles + 7) / 8;            // 8 waves / 256-thr block

    // ---- layer 1 (K=3)
    zero_kernel<<<(n3 + 255) / 256, 256, 0, stream>>>(bufC, n3);
    scatter3_kernel<<<(NEDGES + 255) / 256, 256, 0, stream>>>(x, src, dst, bufC);
    layer1_kernel<<<(n64 + 255) / 256, 256, 0, stream>>>(
        x, bufC, Wc1, bc1, W11, b11, W12, b12, W13, b13, bufA);

    // ---- layer 2 (WMMA)
    zero_kernel<<<(n64 + 255) / 256, 256, 0, stream>>>(bufC, n64);
    scatter64_kernel<<<(int)((e64 + 255) / 256), 256, 0, stream>>>(bufA, src, dst, bufC);
    layer_wmma_kernel<<<wmmaBlocks, 256, 0, stream>>>(
        bufA, bufC, W21, b21, Wc2, bc2, W22, b22, W23, b23, bufB);

    // ---- layer 3 (WMMA)
    zero_kernel<<<(n64 + 255) / 256, 256, 0, stream>>>(bufC, n64);
    scatter64_kernel<<<(int)((e64 + 255) / 256), 256, 0, stream>>>(bufB, src, dst, bufC);
    layer_wmma_kernel<<<wmmaBlocks, 256, 0, stream>>>(
        bufB, bufC, W31, b31, Wc3, bc3, W32, b32, W33, b33, bufA);

    // ---- pool + BN + head
    pool_bn_kernel<<<NGRAPH, 64, 0, stream>>>(bufA, bn_g, bn_b, bn_m, bn_v, pooled);
    head_kernel<<<NGRAPH, 32, 0, stream>>>(pooled, Wf1, bf1, Wf2, bf2, out);
}